// SphericalHarmonics_71683004170705
// MI455X (gfx1250) — compile-verified
//
#include <hip/hip_runtime.h>

#define LMAXX 10
#define NC 121              // (LMAX+1)^2
#define PTS_PER_BLOCK 128
#define WAVES_PER_BLOCK 4
#define TILE_FLOATS (32 * NC)   // 3872 floats = 15488 bytes per wave

typedef __attribute__((ext_vector_type(4))) float    f32x4;
typedef __attribute__((ext_vector_type(4))) unsigned u32x4;
typedef __attribute__((ext_vector_type(8))) int      i32x8;
typedef __attribute__((ext_vector_type(4))) int      i32x4;

// ---- compile-time sqrt (Newton) so K_lm table is baked into the binary ----
constexpr double csqrt(double x) {
  double g = x > 1.0 ? x : 1.0;
  for (int i = 0; i < 64; ++i) g = 0.5 * (g + x / g);
  return g;
}

struct KTab {
  float v[LMAXX + 1][LMAXX + 1];
  constexpr KTab() : v{} {
    for (int l = 0; l <= LMAXX; ++l) {
      for (int m = 0; m <= l; ++m) {
        double ratio = 1.0;                       // (l-m)!/(l+m)!
        for (int t = l - m + 1; t <= l + m; ++t) ratio /= (double)t;
        double k = csqrt((2.0 * l + 1.0) * ratio);
        if (m > 0) k *= csqrt(2.0);
        v[l][m] = (float)k;
      }
    }
  }
};
constexpr KTab kK{};

constexpr float kRcp[LMAXX + 1] = {
    0.0f, 1.0f, 0.5f, 1.0f / 3.0f, 0.25f, 0.2f,
    1.0f / 6.0f, 1.0f / 7.0f, 0.125f, 1.0f / 9.0f, 0.1f};

__global__ __launch_bounds__(PTS_PER_BLOCK) void sh_kernel(
    const float* __restrict__ lonlatr, float* __restrict__ out, int n) {
  __shared__ float tile[WAVES_PER_BLOCK][TILE_FLOATS];

  const int lane = threadIdx.x & 31;
  const int wv   = threadIdx.x >> 5;
  const int waveBase = blockIdx.x * PTS_PER_BLOCK + wv * 32;
  if (waveBase >= n) return;          // N % 32 == 0 -> whole wave in/out
  const int i = waveBase + lane;

  const float lon = lonlatr[3 * i + 0];
  const float lat = lonlatr[3 * i + 1];
  const float r   = lonlatr[3 * i + 2];

  constexpr float DEG  = 0.017453292519943295f;
  constexpr float PI_F = 3.14159265358979323846f;
  float phi   = (lon + 180.0f) * DEG;
  float theta = (90.0f - lat) * DEG;
  theta = fminf(fmaxf(theta, 1e-7f), PI_F - 1e-7f);
  float x = __cosf(theta);
  x = fminf(fmaxf(x, -1.0f + 1e-7f), 1.0f - 1e-7f);
  const float somx2 = sqrtf(fmaxf((1.0f - x) * (1.0f + x), 0.0f));
  const float cphi = __cosf(phi);
  const float sphi = __sinf(phi);
  const float rr = 6371000.0f / r;

  float rad[LMAXX + 1];
  rad[0] = 1.0f;
#pragma unroll
  for (int l = 1; l <= LMAXX; ++l) rad[l] = rad[l - 1] * rr;

  // stride 121 (odd) across lanes -> bank-conflict-free LDS scatter
  float* __restrict__ myrow = &tile[wv][lane * NC];

  float cm = 1.0f, sm = 0.0f;   // cos(m*phi), sin(m*phi)
  float pmm = 1.0f;             // P_mm with Condon-Shortley phase
#pragma unroll
  for (int m = 0; m <= LMAXX; ++m) {
    float p0 = pmm;                              // P_{m,m}
    float p1 = x * (float)(2 * m + 1) * p0;      // P_{m+1,m}
#pragma unroll
    for (int l = m; l <= LMAXX; ++l) {
      float P;
      if (l == m) {
        P = p0;
      } else if (l == m + 1) {
        P = p1;
      } else {
        float pn = ((float)(2 * l - 1) * x * p1 - (float)(l + m - 1) * p0) *
                   kRcp[l - m];
        p0 = p1; p1 = pn; P = pn;
      }
      const float base = kK.v[l][m] * P * rad[l];
      const int c = l * l + l;
      if (m == 0) {
        myrow[c] = base;
      } else {
        myrow[c + m] = base * cm;
        myrow[c - m] = base * sm;
      }
    }
    const float cn = cm * cphi - sm * sphi;
    const float sn = sm * cphi + cm * sphi;
    cm = cn; sm = sn;
    pmm *= -(float)(2 * m + 1) * somx2;
  }

  // Order all DS writes before the tile is consumed (TDM or VMEM readback),
  // and keep the LDS stores from being eliminated.
  asm volatile("s_wait_dscnt 0" ::: "memory");

  float* __restrict__ dst = out + (size_t)waveBase * NC;

#if __has_builtin(__builtin_amdgcn_tensor_store_from_lds) && \
    __has_builtin(__builtin_amdgcn_s_wait_tensorcnt)
  {
    // D# group 0: count=1 | lds_addr | global_addr[56:0] | type=2
    const unsigned ldsaddr = (unsigned)(size_t)(void*)&tile[wv][0];
    const unsigned long long ga = (unsigned long long)(size_t)dst;
    u32x4 g0;
    g0[0] = 1u;                                   // count = 1 (valid)
    g0[1] = ldsaddr;                              // LDS byte address
    g0[2] = (unsigned)(ga & 0xFFFFFFFFull);       // global_addr[31:0]
    g0[3] = (unsigned)((ga >> 32) & 0x01FFFFFFull) | 0x80000000u; // [56:32] | type=2

    // D# group 1: 1-D tile of 3872 fp32 elements, contiguous
    i32x8 g1;
    g1[0] = 0x00020000;            // workgroup_mask=0, data_size=2 (4B)
    g1[1] = (int)(3872u << 16);    // tensor_dim0[15:0] @ bits 63:48
    g1[2] = (int)(1u << 16);       // tensor_dim0[31:16]=0, tensor_dim1=1
    g1[3] = (int)(3872u << 16);    // tensor_dim1 hi=0, tile_dim0=3872
    g1[4] = 1;                     // tile_dim1=1, tile_dim2=0
    g1[5] = 3872;                  // tensor_dim0_stride[31:0]
    g1[6] = 0;                     // stride hi / dim1 stride lo
    g1[7] = 0;

    i32x4 gz = {0, 0, 0, 0};
#if __clang_major__ >= 23
    i32x8 gz8 = {0, 0, 0, 0, 0, 0, 0, 0};
    __builtin_amdgcn_tensor_store_from_lds(g0, g1, gz, gz, gz8, 0);
#else
    __builtin_amdgcn_tensor_store_from_lds(g0, g1, gz, gz, 0);
#endif
    __builtin_amdgcn_s_wait_tensorcnt(0);
  }
#else
  // Fallback: fully coalesced 128-bit non-temporal stores.
  {
    const float* __restrict__ src = &tile[wv][0];
#pragma unroll 5
    for (int t = 0; t < 30; ++t) {
      const int o = t * 128 + lane * 4;
      f32x4 v = *(const f32x4*)(src + o);
      __builtin_nontemporal_store(v, (f32x4*)(dst + o));
    }
    // tail: floats 3840..3871, one per lane
    __builtin_nontemporal_store(src[3840 + lane], dst + 3840 + lane);
  }
#endif
}

extern "C" void kernel_launch(void* const* d_in, const int* in_sizes, int n_in,
                              void* d_out, int out_size, void* d_ws, size_t ws_size,
                              hipStream_t stream) {
  const float* lonlatr = (const float*)d_in[0];
  float* out = (float*)d_out;
  const int n = in_sizes[0] / 3;
  const int blocks = (n + PTS_PER_BLOCK - 1) / PTS_PER_BLOCK;
  sh_kernel<<<dim3(blocks), dim3(PTS_PER_BLOCK), 0, stream>>>(lonlatr, out, n);
}